// HybridQuanvLayer_65266323030223
// MI455X (gfx1250) — compile-verified
//
#include <hip/hip_runtime.h>
#include <hip/hip_bf16.h>

typedef __attribute__((ext_vector_type(2))) float v2f;
typedef __attribute__((ext_vector_type(8))) float v8f;

struct WPtrs { const float* p[14]; };

// ---------------- complex helpers (float2 = re,im) ----------------
__device__ __forceinline__ float2 cmul(float2 a, float2 b) {
    return make_float2(a.x * b.x - a.y * b.y, a.x * b.y + a.y * b.x);
}
__device__ __forceinline__ float2 cadd(float2 a, float2 b) {
    return make_float2(a.x + b.x, a.y + b.y);
}

// Left-multiply U (16x16, in LDS) by embedded 1-qubit gate g on `wire`
// (wire 0 most significant -> row mask 8>>wire). Thread (i,j) owns U[i][j].
__device__ __forceinline__ void apply1q(float2* U, int i, int j, int wire,
                                        float2 g00, float2 g01,
                                        float2 g10, float2 g11) {
    int m  = 8 >> wire;
    int ip = i ^ m;
    float2 a = U[i * 16 + j];
    float2 b = U[ip * 16 + j];
    __syncthreads();
    float2 r;
    if (i & m) r = cadd(cmul(g10, b), cmul(g11, a));
    else       r = cadd(cmul(g00, a), cmul(g01, b));
    U[i * 16 + j] = r;
    __syncthreads();
}

__device__ __forceinline__ void apply_u3(float2* U, int i, int j, int wire,
                                         float th, float ph, float lm) {
    float ct = __cosf(0.5f * th), st = __sinf(0.5f * th);
    float2 ep = make_float2(__cosf(ph), __sinf(ph));
    float2 el = make_float2(__cosf(lm), __sinf(lm));
    float2 g00 = make_float2(ct, 0.f);
    float2 g01 = make_float2(-el.x * st, -el.y * st);
    float2 g10 = make_float2(ep.x * st, ep.y * st);
    float2 epel = cmul(ep, el);
    float2 g11 = make_float2(epel.x * ct, epel.y * ct);
    apply1q(U, i, j, wire, g00, g01, g10, g11);
}

__device__ __forceinline__ void apply_ry(float2* U, int i, int j, int wire, float t) {
    float c = __cosf(0.5f * t), s = __sinf(0.5f * t);
    apply1q(U, i, j, wire,
            make_float2(c, 0.f), make_float2(-s, 0.f),
            make_float2(s, 0.f), make_float2(c, 0.f));
}

__device__ __forceinline__ void apply_rz(float2* U, int i, int j, int wire, float t) {
    float c = __cosf(0.5f * t), s = __sinf(0.5f * t);
    apply1q(U, i, j, wire,
            make_float2(c, -s), make_float2(0.f, 0.f),
            make_float2(0.f, 0.f), make_float2(c, s));
}

__device__ __forceinline__ void apply_cnot(float2* U, int i, int j, int c, int t) {
    int mc = 8 >> c, mt = 8 >> t;
    float2 a = U[i * 16 + j];
    float2 b = U[(i ^ mt) * 16 + j];
    __syncthreads();
    U[i * 16 + j] = (i & mc) ? b : a;
    __syncthreads();
}

// ---------------- kernel 1: build Mcomb[16][32] = [Re U[n][k] | Im U[n][k]] --
__global__ __launch_bounds__(256) void build_circuit_matrix(WPtrs wp, float* __restrict__ M) {
    __shared__ float2 U[256];
    __shared__ float  wAll[30];   // packed: per set of 7 arrays -> 15 floats

    const int tid = threadIdx.x;
    const int i = tid >> 4, j = tid & 15;

    if (tid < 14) {
        const int sz[14]  = {3,3,1,1,1,3,3, 3,3,1,1,1,3,3};
        const int off[14] = {0,3,6,7,8,9,12, 15,18,21,22,23,24,27};
        for (int k = 0; k < sz[tid]; ++k) wAll[off[tid] + k] = wp.p[tid][k];
    }
    U[tid] = make_float2((i == j) ? 1.f : 0.f, 0.f);
    __syncthreads();

    const int pa[4] = {0, 1, 2, 3};
    const int pb[4] = {1, 2, 3, 0};
    for (int s = 0; s < 2; ++s) {
        const int o = s * 15;
        for (int q = 0; q < 4; ++q) {
            const int a = pa[q], b = pb[q];
            apply_u3 (U, i, j, a, wAll[o+0],  wAll[o+1],  wAll[o+2]);
            apply_u3 (U, i, j, b, wAll[o+3],  wAll[o+4],  wAll[o+5]);
            apply_cnot(U, i, j, a, b);
            apply_ry (U, i, j, a, wAll[o+6]);
            apply_rz (U, i, j, b, wAll[o+7]);
            apply_cnot(U, i, j, b, a);
            apply_ry (U, i, j, a, wAll[o+8]);
            apply_cnot(U, i, j, a, b);
            apply_u3 (U, i, j, a, wAll[o+9],  wAll[o+10], wAll[o+11]);
            apply_u3 (U, i, j, b, wAll[o+12], wAll[o+13], wAll[o+14]);
        }
    }

    // psi[b][j] = sum_k psi0[b][k] * U[j][k]
    // Store Mcomb[k][n] = Re U[n][k] (cols 0-15), Im U[n][k] (cols 16-31).
    float2 u = U[i * 16 + j];        // U[row i][col j]
    M[j * 32 + i]      = u.x;        // Mcomb[k=j][n=i]
    M[j * 32 + 16 + i] = u.y;
}

// ---------------- kernel 2: WMMA batch evaluation (transposed GEMM) ---------
// D[state j][patch b] = sum_K  U-plane[j][K] * psi0[b][K]
//   A (16x4 per chunk) = circuit matrix rows  (wave-constant, loaded once)
//   B (4x16 per chunk) = psi0 components      (per-patch, rebuilt per tile)
// D layout: lane b holds states j=0..7 (VGPRs 0..7); lane b+16 holds j=8..15.
__global__ __launch_bounds__(256) void quanv_forward(const float* __restrict__ patches,
                                                     const float* __restrict__ M,
                                                     float* __restrict__ out, int B) {
    const int lane = threadIdx.x & 31;
    const int wid  = threadIdx.x >> 5;
    const int h = lane >> 4;       // lane half: selects K sub-pair / state group
    const int n = lane & 15;       // A: row j ; B/D: patch column b
    const int gwave  = blockIdx.x * (blockDim.x >> 5) + wid;
    const int nwaves = gridDim.x * (blockDim.x >> 5);
    const int ntiles = (B + 15) >> 4;

    // A operands: chunk cc covers K = 4cc..4cc+3.
    // Lane holds row j=n; .x -> K = 4cc+2h, .y -> K = 4cc+2h+1.
    v2f ar[4], ai[4];
#pragma unroll
    for (int cc = 0; cc < 4; ++cc) {
        int k0 = 4 * cc + 2 * h;
        ar[cc].x = M[k0 * 32 + n];            // Re U[j=n][k0]
        ar[cc].y = M[(k0 + 1) * 32 + n];
        ai[cc].x = M[k0 * 32 + 16 + n];       // Im U[j=n][k0]
        ai[cc].y = M[(k0 + 1) * 32 + 16 + n];
    }

    const float HPI = 1.57079632679489662f;   // pi/2

    for (int tile = gwave; tile < ntiles; tile += nwaves) {
        const int p = tile * 16 + n;           // both halves load the same 16 patches
        float4 x = make_float4(0.f, 0.f, 0.f, 0.f);
        if (p < B) x = reinterpret_cast<const float4*>(patches)[p];

        // half-angle = (pi/2)*sigmoid(x); v_w = (cos, sin)
        float h0 = HPI / (1.f + __expf(-x.x));
        float h1 = HPI / (1.f + __expf(-x.y));
        float h2 = HPI / (1.f + __expf(-x.z));
        float h3 = HPI / (1.f + __expf(-x.w));
        float c0 = __cosf(h0), s0 = __sinf(h0);
        float c1 = __cosf(h1), s1 = __sinf(h1);
        float c2 = __cosf(h2), s2 = __sinf(h2);
        float c3 = __cosf(h3), s3 = __sinf(h3);

        // psi0[K], K bits (b0 b1 b2 b3): b0b1 = chunk cc, b2 = half h, b3 = element
        float t01[4] = { c0 * c1, c0 * s1, s0 * c1, s0 * s1 };
        float w2 = h ? s2 : c2;
        float q0 = w2 * c3, q1 = w2 * s3;

        v8f dr = {};   // Re(psi): states x patches
        v8f di = {};   // Im(psi)
#pragma unroll
        for (int cc = 0; cc < 4; ++cc) {
            v2f bv; bv.x = t01[cc] * q0; bv.y = t01[cc] * q1;
            dr = __builtin_amdgcn_wmma_f32_16x16x4_f32(false, ar[cc], false, bv,
                                                       (short)0, dr, false, false);
            di = __builtin_amdgcn_wmma_f32_16x16x4_f32(false, ai[cc], false, bv,
                                                       (short)0, di, false, false);
        }

        // probs for states j = r + 8h of patch b = n, all in this lane's registers
        float pr[8];
#pragma unroll
        for (int r = 0; r < 8; ++r) pr[r] = dr[r] * dr[r] + di[r] * di[r];

        // Signed partial sums, signs are compile-time in r (j = r + 8h):
        //   k=0: sign = bit3(j) = h        k=1: bit2(j) = (r>>2)&1
        //   k=2: bit1(j) = (r>>1)&1        k=3: bit0(j) = r&1
        float a0 = pr[0] + pr[1], a1 = pr[2] + pr[3];
        float a2 = pr[4] + pr[5], a3 = pr[6] + pr[7];
        float t3 = (pr[0] - pr[1]) + (pr[2] - pr[3]) + (pr[4] - pr[5]) + (pr[6] - pr[7]);
        float t2 = (a0 - a1) + (a2 - a3);
        float c01 = a0 + a1, c23 = a2 + a3;
        float t1 = c01 - c23;
        float sum = c01 + c23;
        float t0 = h ? -sum : sum;

        // combine the two state-halves (lane b <-> lane b+16)
        t0 += __shfl_xor(t0, 16, 32);
        t1 += __shfl_xor(t1, 16, 32);
        t2 += __shfl_xor(t2, 16, 32);
        t3 += __shfl_xor(t3, 16, 32);

        // lanes 0..15 store one patch's 4 expectations as a single float4
        if (h == 0 && p < B) {
            reinterpret_cast<float4*>(out)[p] = make_float4(t0, t1, t2, t3);
        }
    }
}

extern "C" void kernel_launch(void* const* d_in, const int* in_sizes, int n_in,
                              void* d_out, int out_size, void* d_ws, size_t ws_size,
                              hipStream_t stream) {
    (void)n_in; (void)out_size; (void)ws_size;
    const float* patches = (const float*)d_in[0];
    WPtrs wp;
    for (int i = 0; i < 14; ++i) wp.p[i] = (const float*)d_in[1 + i];
    float* M   = (float*)d_ws;          // 16x32 f32 = 2 KB
    float* out = (float*)d_out;
    const int B = in_sizes[0] / 4;

    build_circuit_matrix<<<1, 256, 0, stream>>>(wp, M);

    const int ntiles = (B + 15) >> 4;
    int blocks = (ntiles + 63) / 64;    // ~8 tiles per wave at 8 waves/block
    if (blocks > 1024) blocks = 1024;
    if (blocks < 1)    blocks = 1;
    quanv_forward<<<blocks, 256, 0, stream>>>(patches, M, out, B);
}